// GCNStage4_ReduceSum_41807211659496
// MI455X (gfx1250) — compile-verified
//
#include <hip/hip_runtime.h>

#define FEAT 32

// ---------------------------------------------------------------------------
// Scatter-add edge messages into target node rows.  (Placed FIRST in the
// module so its disassembly appears in the feedback snippet — verifying the
// atomics lower to single no-return global_atomic_add_f32, not a CAS loop.)
//
// Wave32 layout: each wave processes 4 edges per iteration.
//   lane = threadIdx.x & 31
//   esub = lane >> 3        -> which of the wave's 4 edges (0..3)
//   fq   = (lane & 7) * 4   -> feature quad start (0,4,...,28)
// Per iteration each lane does one B128 load of msg (wave: 512B contiguous)
// and 4 no-return global_atomic_add_f32 at device scope (resolved in L2;
// the 12.8 MB output lives entirely in the 192 MB L2).
// ---------------------------------------------------------------------------
__global__ __launch_bounds__(256) void scatter_add_kernel(
    const float* __restrict__ msg,
    const int*   __restrict__ tgt,   // edge_index row 1 (int32)
    float*       __restrict__ out,
    int num_edges) {
  const int lane = threadIdx.x & 31;
  const int gwave = (blockIdx.x * blockDim.x + threadIdx.x) >> 5;
  const int nwave = (gridDim.x * blockDim.x) >> 5;
  const int esub = lane >> 3;         // 0..3
  const int fq   = (lane & 7) << 2;   // 0,4,...,28

  const long long stride = (long long)nwave * 4;

  for (long long base = (long long)gwave * 4; base < num_edges; base += stride) {
    const long long e = base + (long long)esub;

    // Prefetch next sweep's msg row segment into cache (global_prefetch_b8).
    const long long pe = e + stride;
    if (pe < (long long)num_edges) {
      __builtin_prefetch(msg + pe * FEAT + fq, 0, 1);
    }

    if (e < (long long)num_edges) {
      const int t = tgt[e];
      const float4 m = *reinterpret_cast<const float4*>(msg + e * FEAT + fq);
      float* o = out + (long long)t * FEAT + fq;
      // No-return f32 atomics at device scope -> L2-resident RMW.
      (void)__hip_atomic_fetch_add(o + 0, m.x, __ATOMIC_RELAXED, __HIP_MEMORY_SCOPE_AGENT);
      (void)__hip_atomic_fetch_add(o + 1, m.y, __ATOMIC_RELAXED, __HIP_MEMORY_SCOPE_AGENT);
      (void)__hip_atomic_fetch_add(o + 2, m.z, __ATOMIC_RELAXED, __HIP_MEMORY_SCOPE_AGENT);
      (void)__hip_atomic_fetch_add(o + 3, m.w, __ATOMIC_RELAXED, __HIP_MEMORY_SCOPE_AGENT);
    }
  }
}

// ---------------------------------------------------------------------------
// Zero the output accumulator (d_out is poisoned by the harness).
// 128-bit stores, fully coalesced.
// ---------------------------------------------------------------------------
__global__ __launch_bounds__(256) void zero_kernel(float4* __restrict__ out, int n4) {
  int i = blockIdx.x * blockDim.x + threadIdx.x;
  if (i < n4) {
    out[i] = float4{0.0f, 0.0f, 0.0f, 0.0f};
  }
}

// ---------------------------------------------------------------------------
// Host-side launch.
//   d_in[0] = msg        (float32, NUM_EDGES * FEAT)
//   d_in[1] = edge_index (int32,   2 * NUM_EDGES)  -- row 1 holds targets
//   d_in[2] = num_nodes  (scalar; implied by out_size)
// ---------------------------------------------------------------------------
extern "C" void kernel_launch(void* const* d_in, const int* in_sizes, int n_in,
                              void* d_out, int out_size, void* d_ws, size_t ws_size,
                              hipStream_t stream) {
  const float* msg        = (const float*)d_in[0];
  const int*   edge_index = (const int*)d_in[1];
  float*       out        = (float*)d_out;

  const int num_edges = in_sizes[0] / FEAT;      // 1,600,000
  const int* tgt = edge_index + num_edges;       // second row of edge_index

  // Zero the accumulator first (same stream => ordered before scatter).
  const int n4 = out_size / 4;
  zero_kernel<<<(n4 + 255) / 256, 256, 0, stream>>>((float4*)out, n4);

  // Grid-stride scatter: 4096 blocks x 8 waves = 32768 waves,
  // each wave covers 4 edges/iter -> ~12 sweeps over 1.6M edges.
  const int blocks = 4096;
  scatter_add_kernel<<<blocks, 256, 0, stream>>>(msg, tgt, out, num_edges);
}